// CausalTrajectoryPrediction_15023795601871
// MI455X (gfx1250) — compile-verified
//
#include <hip/hip_runtime.h>

typedef __attribute__((ext_vector_type(2))) float v2f;
typedef __attribute__((ext_vector_type(8))) float v8f;

#define N_NODES 256
#define HID     1024
#define MID     256
#define K3      512   // M + N

// Compute 16 dot products (rows h0..h0+15 of W, row stride `rowStride` floats)
// against a vector in LDS of length klen (multiple of 4), using
// V_WMMA_F32_16X16X4_F32 with the vector chunk replicated across all 16 B-columns.
// Returns the per-lane v8f accumulator (all 16 columns hold identical results).
// Prefetches the wave's NEXT tile (8 tiles * 16 rows ahead) one cacheline-step
// at a time while streaming the current one.
__device__ __forceinline__ v8f gemv16_wmma(const float* __restrict__ Wt,
                                           int rowStride,
                                           const float* __restrict__ vlds,
                                           int klen, int lane) {
  const int sub = lane >> 4;   // 0: K=0,1   1: K=2,3  (A-matrix 16x4 layout)
  const int row = lane & 15;   // M index
  const float* wp = Wt + (size_t)row * rowStride + sub * 2;
  const float* vp = vlds + sub * 2;
  const int pfd = rowStride * 128;   // next tile this wave will process
  v8f acc = {0.f, 0.f, 0.f, 0.f, 0.f, 0.f, 0.f, 0.f};
#pragma unroll 8
  for (int c = 0; c < klen; c += 4) {
    if ((c & 31) == 0)                     // once per 128B line step per lane
      __builtin_prefetch(wp + c + pfd, 0, 1);
    v2f a, b;
    a.x = wp[c];      // W[row][c + sub*2]
    a.y = wp[c + 1];  // W[row][c + sub*2 + 1]
    b.x = vp[c];      // v[c + sub*2]     replicated over the 16 B-columns
    b.y = vp[c + 1];  // v[c + sub*2 + 1]
    acc = __builtin_amdgcn_wmma_f32_16x16x4_f32(false, a, false, b,
                                                (short)0, acc, false, false);
  }
  return acc;
}

__global__ __launch_bounds__(256)
void ctp_kernel(const float* __restrict__ x,
                const float* __restrict__ W1,
                const float* __restrict__ W2,
                const float* __restrict__ W3,
                const float* __restrict__ b3,
                const float* __restrict__ W4,
                const float* __restrict__ b4,
                float* __restrict__ out) {
  __shared__ float v1[N_NODES];  // inputs_1 row n
  __shared__ float r1[HID];
  __shared__ float r2[MID];
  __shared__ float hl[HID];
  __shared__ float red[8];

  const int n    = blockIdx.x;
  const int tid  = threadIdx.x;
  const int lane = tid & 31;
  const int wid  = tid >> 5;

  if (tid < N_NODES) {
    float xv = x[tid];
    v1[tid] = (tid == n) ? 0.f : xv;   // x with x_n zeroed
  }
  __syncthreads();
  const float xn = x[n];

  const float* W1n = W1 + (size_t)n * HID * N_NODES;
  const float* W2n = W2 + (size_t)n * MID * HID;
  const float* W3n = W3 + (size_t)n * HID * K3;
  const float* b3n = b3 + (size_t)n * HID;
  const float* W4n = W4 + (size_t)n * HID;

  // ---- Layer 1: r1 = relu(W1n[1024x256] @ v1) ----
  for (int t = wid; t < HID / 16; t += 8) {
    const int h0 = t * 16;
    v8f acc = gemv16_wmma(W1n + (size_t)h0 * N_NODES, N_NODES, v1, N_NODES, lane);
    if ((lane & 15) == 0) {               // lanes 0 (M=0..7) and 16 (M=8..15)
      const int base = h0 + (lane >> 4) * 8;
#pragma unroll
      for (int r = 0; r < 8; ++r) r1[base + r] = fmaxf(acc[r], 0.f);
    }
  }
  __syncthreads();

  // ---- Layer 2: r2 = relu(W2n[256x1024] @ r1) ----
  for (int t = wid; t < MID / 16; t += 8) {
    const int m0 = t * 16;
    v8f acc = gemv16_wmma(W2n + (size_t)m0 * HID, HID, r1, HID, lane);
    if ((lane & 15) == 0) {
      const int base = m0 + (lane >> 4) * 8;
#pragma unroll
      for (int r = 0; r < 8; ++r) r2[base + r] = fmaxf(acc[r], 0.f);
    }
  }
  __syncthreads();

  // ---- Layer 3: hl = relu(W3n[:, :256] @ r2 + W3n[:, 256+n]*xn + b3n) ----
  // (inputs_2 row n has a single nonzero x[n] at position n -> read ONE column
  //  of the [., 256:512] half instead of streaming 256 MB of dead weights)
  for (int t = wid; t < HID / 16; t += 8) {
    const int h0 = t * 16;
    v8f acc = gemv16_wmma(W3n + (size_t)h0 * K3, K3, r2, MID, lane);
    if ((lane & 15) == 0) {
      const int base = h0 + (lane >> 4) * 8;
#pragma unroll
      for (int r = 0; r < 8; ++r) {
        const int h = base + r;
        float v = acc[r] + W3n[(size_t)h * K3 + MID + n] * xn + b3n[h];
        hl[h] = fmaxf(v, 0.f);
      }
    }
  }
  __syncthreads();

  // ---- Layer 4: out[n] = relu(dot(W4n, hl) + b4[n]) ----
  float p = 0.f;
  for (int j = tid; j < HID; j += 256) p += W4n[j] * hl[j];
#pragma unroll
  for (int off = 16; off > 0; off >>= 1) p += __shfl_xor(p, off, 32);
  if (lane == 0) red[wid] = p;
  __syncthreads();
  if (tid == 0) {
    float s = 0.f;
#pragma unroll
    for (int w = 0; w < 8; ++w) s += red[w];
    out[n] = fmaxf(s + b4[n], 0.f);
  }
}

extern "C" void kernel_launch(void* const* d_in, const int* in_sizes, int n_in,
                              void* d_out, int out_size, void* d_ws, size_t ws_size,
                              hipStream_t stream) {
  const float* x  = (const float*)d_in[0];
  const float* W1 = (const float*)d_in[1];
  const float* W2 = (const float*)d_in[2];
  const float* W3 = (const float*)d_in[3];
  const float* b3 = (const float*)d_in[4];
  const float* W4 = (const float*)d_in[5];
  const float* b4 = (const float*)d_in[6];
  // d_in[7] = t (unused by the math)
  float* out = (float*)d_out;
  ctp_kernel<<<dim3(N_NODES), dim3(256), 0, stream>>>(x, W1, W2, W3, b3, W4, b4, out);
}